// GCN_88089779241259
// MI455X (gfx1250) — compile-verified
//
#include <hip/hip_runtime.h>

typedef float v2f __attribute__((ext_vector_type(2)));
typedef float v8f __attribute__((ext_vector_type(8)));

#define BN_EPS 1e-5f

// ---------------- utility ----------------
__global__ void k_fill(float* __restrict__ p, float val, int n) {
  int i = blockIdx.x * blockDim.x + threadIdx.x;
  if (i < n) p[i] = val;
}

// deg[col[e]] += ew[e]   (deg pre-filled with 1.0 = self-loop)
__global__ void k_deg_acc(const int* __restrict__ col, const float* __restrict__ ew,
                          float* __restrict__ deg, int E) {
  int e = blockIdx.x * blockDim.x + threadIdx.x;
  if (e < E) atomicAdd(&deg[col[e]], ew[e]);
}

// in-place deg -> 1/sqrt(deg)
__global__ void k_dinv(float* __restrict__ deg, int n) {
  int i = blockIdx.x * blockDim.x + threadIdx.x;
  if (i < n) deg[i] = rsqrtf(deg[i]);
}

// norm[e] = dinv[row]*ew*dinv[col]
__global__ void k_norm(const int* __restrict__ row, const int* __restrict__ col,
                       const float* __restrict__ ew, const float* __restrict__ dinv,
                       float* __restrict__ nrm, int E) {
  int e = blockIdx.x * blockDim.x + threadIdx.x;
  if (e < E) nrm[e] = dinv[row[e]] * ew[e] * dinv[col[e]];
}

// ---------------- WMMA GEMM: H = X @ W ; AGG = dinv^2 * H + bias ----------------
// One wave per 16x16 output tile. N % 16 == 0, Nc % 16 == 0, K % 4 == 0.
__global__ void k_gemm_wmma(const float* __restrict__ X, const float* __restrict__ W,
                            const float* __restrict__ bias, const float* __restrict__ dinv,
                            float* __restrict__ H, float* __restrict__ AGG,
                            int N, int K, int Nc) {
  int gtid  = blockIdx.x * blockDim.x + threadIdx.x;
  int wave  = gtid >> 5;
  int lane  = threadIdx.x & 31;
  int tilesN = Nc >> 4;
  int m0 = (wave / tilesN) << 4;
  int n0 = (wave % tilesN) << 4;
  if (m0 >= N) return;                  // wave-uniform: EXEC all-ones at WMMA

  int lm = lane & 15;                   // row (A) / col (B,C,D) within tile
  int hi = lane >> 4;                   // lane-half selects K pair / M half

  v8f acc = {};
  const float* xrow = X + (size_t)(m0 + lm) * K;
  for (int k = 0; k < K; k += 4) {
    int ka = k + (hi << 1);             // K offsets {0,1} or {2,3}
    v2f a; a.x = xrow[ka];            a.y = xrow[ka + 1];
    v2f b; b.x = W[(size_t)ka * Nc + n0 + lm];
           b.y = W[(size_t)(ka + 1) * Nc + n0 + lm];
    acc = __builtin_amdgcn_wmma_f32_16x16x4_f32(
        /*neg_a=*/false, a, /*neg_b=*/false, b,
        /*c_mod=*/(short)0, acc, /*reuse_a=*/false, /*reuse_b=*/false);
  }

  float bcol = bias[n0 + lm];
#pragma unroll
  for (int r = 0; r < 8; ++r) {
    int m = m0 + r + (hi << 3);         // VGPR r: M = r (lanes 0-15), r+8 (16-31)
    size_t idx = (size_t)m * Nc + n0 + lm;
    float h = acc[r];
    H[idx] = h;
    float di = dinv[m];
    AGG[idx] = di * di * h + bcol;
  }
}

// ---------------- edge scatter: AGG[col,:] += norm[e] * H[row,:] ----------------
__global__ void k_scatter(const int* __restrict__ row, const int* __restrict__ col,
                          const float* __restrict__ nrm, const float* __restrict__ H,
                          float* __restrict__ AGG, int E, int ncLog2) {
  int tid = blockIdx.x * blockDim.x + threadIdx.x;
  int e = tid >> ncLog2;
  if (e >= E) return;
  int d = tid & ((1 << ncLog2) - 1);
  float v = nrm[e] * H[((size_t)row[e] << ncLog2) + d];
  atomicAdd(&AGG[((size_t)col[e] << ncLog2) + d], v);
}

// ---------------- fused BatchNorm(eval) + ReLU ----------------
__global__ void k_bn_relu(const float* __restrict__ A, const float* __restrict__ g,
                          const float* __restrict__ be, const float* __restrict__ m,
                          const float* __restrict__ v, float* __restrict__ X,
                          int total, int ncLog2) {
  int i = blockIdx.x * blockDim.x + threadIdx.x;
  if (i >= total) return;
  int d = i & ((1 << ncLog2) - 1);
  float s = g[d] * rsqrtf(v[d] + BN_EPS);
  float y = (A[i] - m[d]) * s + be[d];
  X[i] = fmaxf(y, 0.0f);
}

// ---------------- global mean pool (accumulate) ----------------
__global__ void k_pool(const int* __restrict__ batch, const float* __restrict__ H,
                       float* __restrict__ sums, float* __restrict__ cnt, int N) {
  int tid = blockIdx.x * blockDim.x + threadIdx.x;
  int i = tid >> 7;
  if (i >= N) return;
  int d = tid & 127;
  int b = batch[i];
  atomicAdd(&sums[(size_t)b * 128 + d], H[(size_t)i * 128 + d]);
  if (d == 0) atomicAdd(&cnt[b], 1.0f);
}

// ---------------- head: pooled -> FC64(ReLU) -> 1 ----------------
__global__ void k_head(const float* __restrict__ sums, const float* __restrict__ cnt,
                       const float* __restrict__ fcw, const float* __restrict__ fcb,
                       const float* __restrict__ ow, const float* __restrict__ ob,
                       float* __restrict__ out) {
  __shared__ float pooled[128];
  __shared__ float hid[64];
  int g = blockIdx.x;
  int t = threadIdx.x;                                  // 0..63
  float c = fmaxf(cnt[g], 1.0f);
  for (int d = t; d < 128; d += 64) pooled[d] = sums[g * 128 + d] / c;
  __syncthreads();
  float acc = fcb[t];
  for (int k = 0; k < 128; ++k) acc += pooled[k] * fcw[k * 64 + t];
  hid[t] = fmaxf(acc, 0.0f);
  __syncthreads();
  if (t == 0) {
    float o = ob[0];
    for (int j = 0; j < 64; ++j) o += hid[j] * ow[j];
    out[g] = o;
  }
}

static inline int cdiv_ll(long long a, long long b) { return (int)((a + b - 1) / b); }

extern "C" void kernel_launch(void* const* d_in, const int* in_sizes, int n_in,
                              void* d_out, int out_size, void* d_ws, size_t ws_size,
                              hipStream_t stream) {
  (void)n_in; (void)out_size; (void)ws_size;
  const float* x     = (const float*)d_in[0];
  const int*   ei    = (const int*)d_in[1];
  const float* ew    = (const float*)d_in[2];
  const int*   batch = (const int*)d_in[3];
  const float* W[3]  = {(const float*)d_in[4],  (const float*)d_in[10], (const float*)d_in[16]};
  const float* B[3]  = {(const float*)d_in[5],  (const float*)d_in[11], (const float*)d_in[17]};
  const float* G[3]  = {(const float*)d_in[6],  (const float*)d_in[12], (const float*)d_in[18]};
  const float* BE[3] = {(const float*)d_in[7],  (const float*)d_in[13], (const float*)d_in[19]};
  const float* M[3]  = {(const float*)d_in[8],  (const float*)d_in[14], (const float*)d_in[20]};
  const float* V[3]  = {(const float*)d_in[9],  (const float*)d_in[15], (const float*)d_in[21]};
  const float* fcw = (const float*)d_in[22];
  const float* fcb = (const float*)d_in[23];
  const float* ow  = (const float*)d_in[24];
  const float* ob  = (const float*)d_in[25];

  const int N = in_sizes[0] / 64;       // 100000
  const int E = in_sizes[1] / 2;        // 1600000
  const int* row = ei;
  const int* col = ei + E;

  // workspace layout (floats)
  float* ws   = (float*)d_ws;
  float* deg  = ws;                          // [N], becomes dinv in place
  float* nrm  = deg + N;                     // [E]
  float* bufH = nrm + E;                     // [N*128]
  float* bufA = bufH + (size_t)N * 128;      // [N*128]
  float* bufX = bufA + (size_t)N * 128;      // [N*128]
  float* psum = bufX + (size_t)N * 128;      // [64*128]
  float* pcnt = psum + 64 * 128;             // [64]

  // ---- normalization precompute ----
  k_fill   <<<cdiv_ll(N, 256), 256, 0, stream>>>(deg, 1.0f, N);
  k_deg_acc<<<cdiv_ll(E, 256), 256, 0, stream>>>(col, ew, deg, E);
  k_dinv   <<<cdiv_ll(N, 256), 256, 0, stream>>>(deg, N);
  k_norm   <<<cdiv_ll(E, 256), 256, 0, stream>>>(row, col, ew, deg, nrm, E);

  // ---- 3 GCN layers ----
  const int dins[3]  = {64, 32, 64};
  const int douts[3] = {32, 64, 128};
  const int logs[3]  = {5, 6, 7};
  const float* lin = x;
  for (int l = 0; l < 3; ++l) {
    int K = dins[l], Nc = douts[l];
    long long tiles = (long long)(N / 16) * (Nc / 16);
    k_gemm_wmma<<<cdiv_ll(tiles * 32, 256), 256, 0, stream>>>(
        lin, W[l], B[l], deg, bufH, bufA, N, K, Nc);
    k_scatter<<<cdiv_ll((long long)E * Nc, 256), 256, 0, stream>>>(
        row, col, nrm, bufH, bufA, E, logs[l]);
    k_bn_relu<<<cdiv_ll((long long)N * Nc, 256), 256, 0, stream>>>(
        bufA, G[l], BE[l], M[l], V[l], bufX, N * Nc, logs[l]);
    lin = bufX;
  }

  // ---- global mean pool + head ----
  k_fill<<<cdiv_ll(64 * 128 + 64, 256), 256, 0, stream>>>(psum, 0.0f, 64 * 128 + 64);
  k_pool<<<cdiv_ll((long long)N * 128, 256), 256, 0, stream>>>(batch, bufX, psum, pcnt, N);
  k_head<<<64, 64, 0, stream>>>(psum, pcnt, fcw, fcb, ow, ob, (float*)d_out);
}